// EnhancedGNN_11450382811735
// MI455X (gfx1250) — compile-verified
//
#include <hip/hip_runtime.h>
#include <hip/hip_bf16.h>

// ---------------------------------------------------------------------------
// GCN on MI455X (gfx1250). Dense GEMMs use V_WMMA_F32_16X16X4_F32 (wave32,
// 16x16 tile per wave, full fp32 fidelity). The 16-row A panel is staged in
// LDS once per block (all 8 waves share the same M-tile), so the WMMA loop
// reads A via ds_load_b64 and only B via global. Edge scatter uses f32
// global atomics against the L2-resident aggregate buffer.
// ---------------------------------------------------------------------------

typedef __attribute__((ext_vector_type(2))) float v2f;
typedef __attribute__((ext_vector_type(8))) float v8f;

#define F_HID   256
#define F_OUTC  128
#define N_GR    64

// deg[i] = 1 (self loop)
__global__ void k_init_deg(float* deg, int n) {
    int i = blockIdx.x * blockDim.x + threadIdx.x;
    if (i < n) deg[i] = 1.0f;
}

// deg[dst[e]] += 1
__global__ void k_deg_accum(const int* __restrict__ dst, float* deg, int E) {
    int e = blockIdx.x * blockDim.x + threadIdx.x;
    if (e < E) atomicAdd(&deg[dst[e]], 1.0f);
}

// deg -> rsqrt(max(deg,1))
__global__ void k_dinv(float* deg, int n) {
    int i = blockIdx.x * blockDim.x + threadIdx.x;
    if (i < n) deg[i] = rsqrtf(fmaxf(deg[i], 1.0f));
}

// agg[node][f] = bias[f]   (grid.x = n, block = 256)
__global__ void k_init_bias(float* __restrict__ agg, const float* __restrict__ b, int n) {
    int node = blockIdx.x;
    if (node >= n) return;
    agg[(size_t)node * F_HID + threadIdx.x] = b[threadIdx.x];
}

// C[MxN] = A[MxK] @ B[KxN] (+ bias[col]).  One wave per 16x16 tile, 8 waves
// per block covering 8 consecutive N-tiles of the SAME M-tile (requires
// tiles_n % 8 == 0, true for N=256 and N=128). The 16xK A panel (contiguous
// in row-major A) is cooperatively staged into LDS once per block.
//
// V_WMMA_F32_16X16X4_F32 layout:
//   A 16x4 : lanes 0-15 rows M=0..15 holding K={0,1}; lanes 16-31 K={2,3}
//   B 4x16 : VGPR0/1 = rows K={0,1} (lanes 0-15), K={2,3} (lanes 16-31)
//   C/D    : VGPR v = row (v + 8*half), col = lane&15
__global__ __launch_bounds__(256) void k_gemm_wmma(
        const float* __restrict__ A, const float* __restrict__ B,
        float* __restrict__ C, const float* __restrict__ bias,
        int M, int K, int N) {
    extern __shared__ float lds[];                 // 16 * K floats

    const int lane    = threadIdx.x & 31;
    const int wave    = threadIdx.x >> 5;
    const int tiles_n = N >> 4;
    const int tm      = (blockIdx.x * 8) / tiles_n;      // shared by all waves
    const int tile    = blockIdx.x * 8 + wave;
    const int tn      = tile % tiles_n;

    // ---- cooperative stage of the 16xK A panel (contiguous region) ----
    {
        const float4* __restrict__ Ablk = (const float4*)(A + (size_t)tm * 16 * K);
        float4* lds4 = (float4*)lds;
        const int n4 = 4 * K;                       // 16*K floats / 4
        for (int i = threadIdx.x; i < n4; i += 256) lds4[i] = Ablk[i];
    }
    __syncthreads();

    if (tile >= (M >> 4) * tiles_n) return;

    const int half = lane >> 4;      // 0: K={0,1}, 1: K={2,3}
    const int l    = lane & 15;
    const int col  = tn * 16 + l;    // B/D column owned by this lane

    const float* __restrict__ Arow = lds + l * K + 2 * half;       // LDS
    const float* __restrict__ Bp   = B + (size_t)(2 * half) * N + col;

    v8f acc = {};
#pragma unroll 4
    for (int k0 = 0; k0 < K; k0 += 4) {
        v2f a = *(const v2f*)(Arow + k0);            // ds_load_b64
        v2f b;
        b.x = Bp[(size_t)k0 * N];
        b.y = Bp[(size_t)k0 * N + N];
        acc = __builtin_amdgcn_wmma_f32_16x16x4_f32(
                /*neg_a=*/false, a, /*neg_b=*/false, b,
                /*c_mod=*/(short)0, acc, /*reuse_a=*/false, /*reuse_b=*/false);
    }

    const float bval = bias ? bias[col] : 0.0f;
#pragma unroll
    for (int v = 0; v < 8; ++v) {
        int r = tm * 16 + v + 8 * half;
        C[(size_t)r * N + col] = acc[v] + bval;
    }
}

// One wave per edge (E real edges + n self loops). Each lane moves 2 float4
// of the 256-wide row: agg[dst] += T[src] * dinv[src]*dinv[dst].
__global__ __launch_bounds__(256) void k_scatter(
        const float* __restrict__ T, const int* __restrict__ src,
        const int* __restrict__ dst, const float* __restrict__ dinv,
        float* __restrict__ agg, int E, int n) {
    int gwave = (int)((blockIdx.x * (size_t)blockDim.x + threadIdx.x) >> 5);
    int lane  = threadIdx.x & 31;
    int total = E + n;
    if (gwave >= total) return;

    int s, d; float nrm;
    if (gwave < E) {
        s = src[gwave]; d = dst[gwave];
        nrm = dinv[s] * dinv[d];
    } else {
        s = gwave - E; d = s;
        float di = dinv[s];
        nrm = di * di;
    }

    const float4* __restrict__ rowp = (const float4*)(T + (size_t)s * F_HID);
    float* __restrict__ outp = agg + (size_t)d * F_HID;
#pragma unroll
    for (int i = 0; i < 2; ++i) {
        int idx = lane + 32 * i;         // float4 index, 64 per row
        float4 v = rowp[idx];
        float* o = outp + idx * 4;
        atomicAdd(o + 0, v.x * nrm);
        atomicAdd(o + 1, v.y * nrm);
        atomicAdd(o + 2, v.z * nrm);
        atomicAdd(o + 3, v.w * nrm);
    }
}

__global__ void k_relu(float* p, long long n) {
    long long i = blockIdx.x * (long long)blockDim.x + threadIdx.x;
    if (i < n) p[i] = fmaxf(p[i], 0.0f);
}

__global__ void k_zero(float* p, long long n) {
    long long i = blockIdx.x * (long long)blockDim.x + threadIdx.x;
    if (i < n) p[i] = 0.0f;
}

// grid.x = n nodes, block = 256 features. Post-ReLU H >= 0, so bit-pattern
// atomicMax with zero-init matches segment_max + isfinite-guard semantics.
__global__ void k_pool(const float* __restrict__ H, const int* __restrict__ batch,
                       float* __restrict__ sums, unsigned* __restrict__ mx,
                       float* __restrict__ cnt, int n) {
    int node = blockIdx.x;
    if (node >= n) return;
    int f = threadIdx.x;
    int g = batch[node];
    float v = H[(size_t)node * F_HID + f];
    atomicAdd(&sums[g * F_HID + f], v);
    atomicMax(&mx[g * F_HID + f], __float_as_uint(v));
    if (f == 0) atomicAdd(&cnt[g], 1.0f);
}

// gbuf[g] = [mean | max | sum]  (64 x 768)
__global__ void k_build_g(const float* __restrict__ sums, const unsigned* __restrict__ mx,
                          const float* __restrict__ cnt, float* __restrict__ gbuf) {
    int g = blockIdx.x;          // 64
    int f = threadIdx.x;         // 256
    float c = fmaxf(cnt[g], 1.0f);
    float s = sums[g * F_HID + f];
    gbuf[(size_t)g * 768 + f]        = s / c;
    gbuf[(size_t)g * 768 + 256 + f]  = __uint_as_float(mx[g * F_HID + f]);
    gbuf[(size_t)g * 768 + 512 + f]  = s;
}

extern "C" void kernel_launch(void* const* d_in, const int* in_sizes, int n_in,
                              void* d_out, int out_size, void* d_ws, size_t ws_size,
                              hipStream_t stream) {
    const float* x     = (const float*)d_in[0];
    const int*   ei    = (const int*)d_in[1];
    const int*   batch = (const int*)d_in[2];
    // d_in[3] = n_graphs (device scalar); compile-time constant N_GR = 64
    const float* W1  = (const float*)d_in[4];
    const float* b1  = (const float*)d_in[5];
    const float* W2  = (const float*)d_in[6];
    const float* b2  = (const float*)d_in[7];
    const float* Wfc = (const float*)d_in[8];
    const float* bfc = (const float*)d_in[9];

    const int n = in_sizes[2];          // 50000 nodes
    const int E = in_sizes[1] / 2;      // 800000 edges
    const int* src = ei;
    const int* dst = ei + E;

    // ---- workspace carve-out (256B aligned) ----
    char* ws = (char*)d_ws;
    size_t off = 0;
    auto carve = [&](size_t bytes) -> char* {
        char* p = ws + off;
        off = (off + bytes + 255) & ~(size_t)255;
        return p;
    };
    float*    dinv = (float*)carve((size_t)n * sizeof(float));
    float*    bufA = (float*)carve((size_t)n * F_HID * sizeof(float));
    float*    bufB = (float*)carve((size_t)n * F_HID * sizeof(float));
    float*    pool = (float*)carve((size_t)(2 * N_GR * F_HID + N_GR) * sizeof(float));
    float*    sums = pool;
    unsigned* mx   = (unsigned*)(pool + N_GR * F_HID);
    float*    cnt  = pool + 2 * N_GR * F_HID;
    float*    gbuf = (float*)carve((size_t)N_GR * 3 * F_HID * sizeof(float));
    (void)ws_size; (void)n_in;

    const long long nfeat = (long long)n * F_HID;

    // ---- degree / normalization ----
    k_init_deg <<<(n + 255) / 256, 256, 0, stream>>>(dinv, n);
    k_deg_accum<<<(E + 255) / 256, 256, 0, stream>>>(dst, dinv, E);
    k_dinv     <<<(n + 255) / 256, 256, 0, stream>>>(dinv, n);

    const int totalEdges    = E + n;
    const int scatterBlks   = (totalEdges + 7) / 8;      // 8 waves / block
    const int gemmTilesBig  = (n / 16) * (F_HID / 16);
    const size_t ldsBig     = (size_t)16 * F_HID * sizeof(float);      // 16 KB
    const size_t ldsFc      = (size_t)16 * 3 * F_HID * sizeof(float);  // 48 KB

    // ---- layer 1: T = X @ W1 ; agg = b1 + scatter ; relu ----
    k_gemm_wmma<<<(gemmTilesBig + 7) / 8, 256, ldsBig, stream>>>(x, W1, bufA, nullptr, n, F_HID, F_HID);
    k_init_bias<<<n, 256, 0, stream>>>(bufB, b1, n);
    k_scatter  <<<scatterBlks, 256, 0, stream>>>(bufA, src, dst, dinv, bufB, E, n);
    k_relu     <<<(int)((nfeat + 255) / 256), 256, 0, stream>>>(bufB, nfeat);

    // ---- layer 2 ----
    k_gemm_wmma<<<(gemmTilesBig + 7) / 8, 256, ldsBig, stream>>>(bufB, W2, bufA, nullptr, n, F_HID, F_HID);
    k_init_bias<<<n, 256, 0, stream>>>(bufB, b2, n);
    k_scatter  <<<scatterBlks, 256, 0, stream>>>(bufA, src, dst, dinv, bufB, E, n);
    k_relu     <<<(int)((nfeat + 255) / 256), 256, 0, stream>>>(bufB, nfeat);

    // ---- pooling ----
    const long long poolElems = 2 * N_GR * F_HID + N_GR;
    k_zero   <<<(int)((poolElems + 255) / 256), 256, 0, stream>>>(pool, poolElems);
    k_pool   <<<n, 256, 0, stream>>>(bufB, batch, sums, mx, cnt, n);
    k_build_g<<<N_GR, 256, 0, stream>>>(sums, mx, cnt, gbuf);

    // ---- final FC: out = g @ Wfc + bfc  (64 x 768 @ 768 x 128) ----
    const int gemmTilesFc = (N_GR / 16) * (F_OUTC / 16);   // 32 tiles
    k_gemm_wmma<<<(gemmTilesFc + 7) / 8, 256, ldsFc, stream>>>(gbuf, Wfc, (float*)d_out, bfc,
                                                               N_GR, 3 * F_HID, F_OUTC);
    (void)out_size;
}